// ChamferLoss_1580547972175
// MI455X (gfx1250) — compile-verified
//
#include <hip/hip_runtime.h>
#include <hip/hip_bf16.h>

typedef __attribute__((ext_vector_type(2))) float v2f;
typedef __attribute__((ext_vector_type(8))) float v8f;

#define NP   4096
#define NB   16
#define EPSF 1e-6f
#define POSINF_BITS 0x7f800000u

static __device__ __forceinline__ unsigned int umin32(unsigned int a,
                                                      unsigned int b) {
    return a < b ? a : b;   // v_min_u32 / v_min3_u32, no canonicalize
}

// Pure-VALU swap between lane halves (lane i <-> lane i^16), no LDS op.
static __device__ __forceinline__ unsigned int xhalf_swap_u(unsigned int v) {
    return (unsigned int)__builtin_amdgcn_permlanex16(
        (int)v, (int)v, 0x76543210, 0xfedcba98, /*fi=*/true, /*bc=*/false);
}
static __device__ __forceinline__ float xhalf_swap_f(float v) {
    return __int_as_float(__builtin_amdgcn_permlanex16(
        __float_as_int(v), __float_as_int(v),
        0x76543210, 0xfedcba98, true, false));
}

// ---------------------------------------------------------------------------
// Kernel 0: init column-min buffer to +inf (bit pattern). Deterministic.
// ---------------------------------------------------------------------------
__global__ void chamfer_init_kernel(unsigned int* colMin, int n) {
    int i = blockIdx.x * blockDim.x + threadIdx.x;
    if (i < n) colMin[i] = POSINF_BITS;
}

// ---------------------------------------------------------------------------
// Kernel 1: WMMA tile sweep over SQUARED distances.
// One wave handles 16 y-rows (one batch) x all 4096 x-columns (256 tiles).
// sq[i,j] = ||y_i||^2 + ||x_j||^2 - 2*y_i.x_j  via a single
// v_wmma_f32_16x16x4_f32 per tile with the norms folded into A/B/C.
// All min-reductions run in the uint bit domain (== float order for the
// non-negative mins that matter) -> pure v_min_u32/v_min3_u32, no
// canonicalize ops, no per-tile clamp. sqrt/eps applied only at epilogues.
// Column mins merge per-block in LDS (ds_min_u32, branchless, both lane
// halves contribute), then flush once per block with global atomicMin.
// 4-tile unroll + 1-group software pipeline amortizes dscnt/loadcnt waits.
// ---------------------------------------------------------------------------
__global__ void __launch_bounds__(256)
chamfer_tile_kernel(const float* __restrict__ x,   // (16,4096,3)
                    const float* __restrict__ y,   // (16,4096,3)
                    unsigned int* __restrict__ colMin,   // (16,4096) sq bits
                    float* __restrict__ rowSums) {       // (4096) per-wave sums
    __shared__ unsigned int cmShared[NP];   // 16 KB per block

    const int tid   = threadIdx.x;
    const int lane  = tid & 31;
    const int wave  = tid >> 5;
    const int gwave = blockIdx.x * (blockDim.x >> 5) + wave;  // 0..4095

    const int b       = gwave >> 8;          // batch 0..15 (uniform per block)
    const int rowBase = (gwave & 255) << 4;  // y-row block start, 0..4080

    const int  sub = lane & 15;   // M (for A) or N (for B/C/D)
    const bool hi  = lane >= 16;  // lane half selects K-pair / row+8

    // init shared column mins
    #pragma unroll
    for (int i = tid; i < NP; i += 256) cmShared[i] = POSINF_BITS;
    __syncthreads();

    const float* xb = x + (size_t)b * NP * 3;
    const float* yb = y + (size_t)b * NP * 3;

    // ---- A operand: row i = rowBase+sub, A[i,:] = (-2y0, -2y1, -2y2, 1)
    // 32-bit A 16x4 layout: VGPR0 = K0 (lanes 0-15) | K2 (lanes 16-31)
    //                       VGPR1 = K1 (lanes 0-15) | K3 (lanes 16-31)
    const float* yp = yb + (rowBase + sub) * 3;
    float yc0 = yp[0], yc1 = yp[1], yc2 = yp[2];
    float y2n = yc0 * yc0 + yc1 * yc1 + yc2 * yc2;

    v2f a;
    a.x = hi ? (-2.0f * yc2) : (-2.0f * yc0);
    a.y = hi ? 1.0f          : (-2.0f * yc1);

    // ---- C init: C[v][lane] = ||y_{rowBase + v + (hi?8:0)}||^2
    v8f cinit;
    #pragma unroll
    for (int v = 0; v < 8; ++v) {
        int src = v + (hi ? 8 : 0);
        cinit[v] = __shfl(y2n, src, 32);
    }

    unsigned int rowMinU[8];
    #pragma unroll
    for (int v = 0; v < 8; ++v) rowMinU[v] = POSINF_BITS;

    // One 16x16 tile: build B operand, WMMA, fold uint mins, ds_min columns.
    // Both lane halves push their own partial (rows 0-7 / 8-15) min: LDS
    // merges them -> no cross-lane op, no exec branch in the loop.
    auto process = [&](int j0, float xc0, float xc1, float xc2) {
        float xn = xc0 * xc0 + xc1 * xc1 + xc2 * xc2;

        // B 4x16 layout mirrors A: VGPR0 = K0|K2, VGPR1 = K1|K3, lanes = N
        v2f bm;
        bm.x = hi ? xc2 : xc0;
        bm.y = hi ? xn  : xc1;

        v8f sq = __builtin_amdgcn_wmma_f32_16x16x4_f32(
            false, a, false, bm, (short)0, cinit, false, false);

        unsigned int u0 = __float_as_uint(sq[0]);
        unsigned int tileColMinU = u0;
        rowMinU[0] = umin32(rowMinU[0], u0);
        #pragma unroll
        for (int v = 1; v < 8; ++v) {
            unsigned int u = __float_as_uint(sq[v]);
            rowMinU[v]  = umin32(rowMinU[v], u);
            tileColMinU = umin32(tileColMinU, u);
        }

        atomicMin(&cmShared[j0 + sub], tileColMinU);   // ds_min_u32
    };

    // ---- software-pipelined sweep, 4 tiles (64 columns) per iteration.
    // Loads for the next 4-tile group are issued before processing the
    // current group; marching pointer keeps immediate load offsets.
    const float* pc = xb + sub * 3;
    float c[4][3];
    #pragma unroll
    for (int t = 0; t < 4; ++t) {
        c[t][0] = pc[t * 48 + 0];
        c[t][1] = pc[t * 48 + 1];
        c[t][2] = pc[t * 48 + 2];
    }
    int j0 = 0;
    for (; j0 < NP - 64; j0 += 64) {
        float n[4][3];
        #pragma unroll
        for (int t = 0; t < 4; ++t) {
            n[t][0] = pc[192 + t * 48 + 0];
            n[t][1] = pc[192 + t * 48 + 1];
            n[t][2] = pc[192 + t * 48 + 2];
        }

        #pragma unroll
        for (int t = 0; t < 4; ++t)
            process(j0 + t * 16, c[t][0], c[t][1], c[t][2]);

        #pragma unroll
        for (int t = 0; t < 4; ++t) {
            c[t][0] = n[t][0]; c[t][1] = n[t][1]; c[t][2] = n[t][2];
        }
        pc += 192;
    }
    #pragma unroll
    for (int t = 0; t < 4; ++t)        // peeled last group (no next loads)
        process(j0 + t * 16, c[t][0], c[t][1], c[t][2]);

    // ---- per-row mins: reduce across the 16 lanes of each half (uint domain)
    #pragma unroll
    for (int v = 0; v < 8; ++v) {
        #pragma unroll
        for (int off = 1; off < 16; off <<= 1)
            rowMinU[v] = umin32(rowMinU[v],
                                (unsigned int)__shfl_xor((int)rowMinU[v], off, 32));
    }
    // lanes 0-15 hold row-min(sq) of row rowBase+v; lanes 16-31 of rowBase+v+8.
    // Apply the deferred clamp + eps + sqrt here (16 sqrts per wave total).
    float localSum = 0.0f;
    #pragma unroll
    for (int v = 0; v < 8; ++v)
        localSum += __builtin_sqrtf(
            EPSF + fmaxf(__uint_as_float(rowMinU[v]), 0.0f));
    localSum += xhalf_swap_f(localSum);         // all 16 rows of this block

    if (lane == 0) rowSums[gwave] = localSum;   // exclusive slot, no atomics

    // ---- flush block-merged column mins: one global atomic per column
    __syncthreads();
    unsigned int* cmB = colMin + b * NP;
    #pragma unroll
    for (int cIdx = tid; cIdx < NP; cIdx += 256)
        atomicMin(&cmB[cIdx], cmShared[cIdx]);
}

// ---------------------------------------------------------------------------
// Kernel 2: deterministic final reduction.
// colMin holds squared-min bit patterns -> apply clamp + eps + sqrt here.
// out = mean(min1) + mean(min2) over 16*4096 points each.
// ---------------------------------------------------------------------------
__global__ void chamfer_reduce_kernel(const unsigned int* __restrict__ colMin,
                                      const float* __restrict__ rowSums,
                                      float* __restrict__ out) {
    __shared__ float smem[256];
    const int tid = threadIdx.x;

    float s = 0.0f;
    for (int i = tid; i < NB * NP; i += 256)
        s += __builtin_sqrtf(EPSF + fmaxf(__uint_as_float(colMin[i]), 0.0f));
    for (int i = tid; i < NB * NP / 16; i += 256)
        s += rowSums[i];

    smem[tid] = s;
    __syncthreads();
    for (int stride = 128; stride > 0; stride >>= 1) {
        if (tid < stride) smem[tid] += smem[tid + stride];
        __syncthreads();
    }
    if (tid == 0) out[0] = smem[0] * (1.0f / (float)(NB * NP));
}

// ---------------------------------------------------------------------------
extern "C" void kernel_launch(void* const* d_in, const int* in_sizes, int n_in,
                              void* d_out, int out_size, void* d_ws, size_t ws_size,
                              hipStream_t stream) {
    const float* x = (const float*)d_in[0];   // (16,4096,3)
    const float* y = (const float*)d_in[1];   // (16,4096,3)
    float* out = (float*)d_out;

    // workspace layout
    unsigned int* colMin  = (unsigned int*)d_ws;                 // 16*4096 u32
    float*        rowSums = (float*)((char*)d_ws + (size_t)NB * NP * 4); // 4096 f32

    // 0) init col-min buffer to +inf
    chamfer_init_kernel<<<dim3((NB * NP + 255) / 256), dim3(256), 0, stream>>>(
        colMin, NB * NP);

    // 1) WMMA tile sweep: 4096 waves, 8 waves/block
    chamfer_tile_kernel<<<dim3(512), dim3(256), 0, stream>>>(
        x, y, colMin, rowSums);

    // 2) deterministic reduction to scalar
    chamfer_reduce_kernel<<<dim3(1), dim3(256), 0, stream>>>(
        colMin, rowSums, out);
}